// GloveTextRNN_19413252178301
// MI455X (gfx1250) — compile-verified
//
#include <hip/hip_runtime.h>
#include <hip/hip_bf16.h>

typedef __attribute__((ext_vector_type(2))) float v2f;
typedef __attribute__((ext_vector_type(8))) float v8f;

// B=8192, T=50, I=50, H=10, L=3, C=5
// Layer-0 projection GEMM: R = B*T = 409600 rows, K = I = 50, N = H = 10.

// ---------------------------------------------------------------------------
// Kernel A: xW0[r, h] = sum_i x[r, i] * W_ih0[h, i] + (b_ih0[h] + b_hh0[h])
// One wave computes a 16(M) x 16(N) tile with V_WMMA_F32_16X16X4_F32,
// looping K in steps of 4 (13 iterations, zero-padded past K=50).
// B fragments come from a zero-padded LDS tile -> unconditional ds_load_b64,
// no exec-mask branching in the WMMA chain.
// ---------------------------------------------------------------------------
#define SB_STRIDE 58   // even (keeps 8B alignment for k even); 58*lm mod 64 distinct for lm 0..15

__global__ __launch_bounds__(256) void rnn_l0_proj_wmma(
    const float* __restrict__ x,      // [409600, 50]
    const float* __restrict__ Wih0,   // [10, 50]
    const float* __restrict__ bih0,   // [10]
    const float* __restrict__ bhh0,   // [10]
    float* __restrict__ xw0)          // [409600, 10]
{
    __shared__ float sB[16 * SB_STRIDE];  // sB[n*SB_STRIDE + k] = B[k][n] = Wih0[n][k], zero-padded
    __shared__ float sBias[16];

    const int tid = threadIdx.x;
    for (int i = tid; i < 16 * SB_STRIDE; i += 256) sB[i] = 0.0f;
    __syncthreads();
    for (int i = tid; i < 500; i += 256) {        // 10 rows x 50 cols
        const int n = i / 50, k = i - n * 50;
        sB[n * SB_STRIDE + k] = Wih0[i];
    }
    if (tid < 16) sBias[tid] = (tid < 10) ? (bih0[tid] + bhh0[tid]) : 0.0f;
    __syncthreads();

    const int lane = tid & 31;
    const int wave = tid >> 5;
    const int tile = blockIdx.x * 8 + wave;       // 25600 tiles total
    const int row0 = tile * 16;
    const int half = lane >> 4;                   // 0: lanes 0-15, 1: lanes 16-31
    const int lm   = lane & 15;

    // A-matrix 16x4 f32 layout: lanes 0-15 hold (K=k0, k0+1), lanes 16-31 hold (K=k0+2, k0+3)
    const float* xrow = x + (row0 + lm) * 50;     // (row0+lm)*50 is even -> 8B aligned
    const float* brow = sB + lm * SB_STRIDE;

    v8f c = {};
    #pragma unroll
    for (int k0 = 0; k0 < 52; k0 += 4) {
        const int k  = k0 + half * 2;             // even
        const int kc = (k > 48) ? 48 : k;         // clamp so the b64 load stays in-bounds
        const v2f xv = *(const v2f*)(xrow + kc);
        v2f a;
        a.x = (k     < 50) ? xv.x : 0.0f;         // constant-folds except last iteration
        a.y = (k + 1 < 50) ? xv.y : 0.0f;
        const v2f b = *(const v2f*)(brow + k);    // zero-padded in LDS, unconditional
        // 8 args: (neg_a, A, neg_b, B, c_mod, C, reuse_a, reuse_b)
        c = __builtin_amdgcn_wmma_f32_16x16x4_f32(
                false, a, false, b, (short)0, c, false, false);
    }

    const float bias = sBias[lm];
    // C/D layout: lanes 0-15: VGPR r = (M=r, N=lm); lanes 16-31: VGPR r = (M=8+r, N=lm)
    if (lm < 10) {
        #pragma unroll
        for (int r = 0; r < 8; ++r) {
            const int m = row0 + half * 8 + r;
            xw0[m * 10 + lm] = c[r] + bias;
        }
    }
}

// ---------------------------------------------------------------------------
// Kernel B: fused 3-layer recurrence + ReLU + FC. One thread per batch row.
// All small weights staged in LDS (broadcast reads, conflict-free).
// ---------------------------------------------------------------------------
__global__ __launch_bounds__(256) void rnn_recurrent_fused(
    const float* __restrict__ xw0,   // [8192*50, 10] precomputed layer-0 input proj (+bias)
    const float* __restrict__ h0,    // [3, 8192, 10]
    const float* __restrict__ Whh0,  // [10,10]
    const float* __restrict__ Wih1, const float* __restrict__ Whh1,
    const float* __restrict__ bih1, const float* __restrict__ bhh1,
    const float* __restrict__ Wih2, const float* __restrict__ Whh2,
    const float* __restrict__ bih2, const float* __restrict__ bhh2,
    const float* __restrict__ fcw,   // [5, 500]
    const float* __restrict__ fcb,   // [5]
    float* __restrict__ out)         // [8192, 5]
{
    __shared__ float sW[3032];
    float* sWhh0 = sW;            // 100
    float* sWih1 = sW + 100;      // 100
    float* sWhh1 = sW + 200;      // 100
    float* sWih2 = sW + 300;      // 100
    float* sWhh2 = sW + 400;      // 100
    float* sb1   = sW + 500;      // 10  (b_ih1 + b_hh1)
    float* sb2   = sW + 510;      // 10  (b_ih2 + b_hh2)
    float* sfcw  = sW + 520;      // 2500
    float* sfcb  = sW + 3020;     // 5

    const int tid = threadIdx.x;
    for (int i = tid; i < 100; i += 256) {
        sWhh0[i] = Whh0[i];
        sWih1[i] = Wih1[i];
        sWhh1[i] = Whh1[i];
        sWih2[i] = Wih2[i];
        sWhh2[i] = Whh2[i];
    }
    if (tid < 10) { sb1[tid] = bih1[tid] + bhh1[tid]; sb2[tid] = bih2[tid] + bhh2[tid]; }
    for (int i = tid; i < 2500; i += 256) sfcw[i] = fcw[i];
    if (tid < 5) sfcb[tid] = fcb[tid];
    __syncthreads();

    const int b = blockIdx.x * 256 + tid;   // 8192 threads

    float hA[10], hB[10], hC[10];
    #pragma unroll
    for (int j = 0; j < 10; ++j) {
        hA[j] = h0[            b * 10 + j];
        hB[j] = h0[ 81920 + b * 10 + j];
        hC[j] = h0[163840 + b * 10 + j];
    }
    float acc[5];
    #pragma unroll
    for (int cc = 0; cc < 5; ++cc) acc[cc] = sfcb[cc];

    const v2f* xp2 = (const v2f*)(xw0 + (size_t)b * 500);   // 8B-aligned rows

    for (int t = 0; t < 50; ++t) {
        float xw[10], hn[10];
        #pragma unroll
        for (int p = 0; p < 5; ++p) {          // 5 x global_load_b64
            const v2f v = xp2[t * 5 + p];
            xw[2 * p]     = v.x;
            xw[2 * p + 1] = v.y;
        }
        // layer 0: h = tanh(xw0_t + hA @ Whh0^T)
        #pragma unroll
        for (int j = 0; j < 10; ++j) {
            float s = xw[j];
            #pragma unroll
            for (int k = 0; k < 10; ++k) s = fmaf(hA[k], sWhh0[j * 10 + k], s);
            hn[j] = tanhf(s);
        }
        #pragma unroll
        for (int j = 0; j < 10; ++j) hA[j] = hn[j];
        // layer 1
        #pragma unroll
        for (int j = 0; j < 10; ++j) {
            float s = sb1[j];
            #pragma unroll
            for (int i = 0; i < 10; ++i) s = fmaf(hA[i], sWih1[j * 10 + i], s);
            #pragma unroll
            for (int k = 0; k < 10; ++k) s = fmaf(hB[k], sWhh1[j * 10 + k], s);
            hn[j] = tanhf(s);
        }
        #pragma unroll
        for (int j = 0; j < 10; ++j) hB[j] = hn[j];
        // layer 2
        #pragma unroll
        for (int j = 0; j < 10; ++j) {
            float s = sb2[j];
            #pragma unroll
            for (int i = 0; i < 10; ++i) s = fmaf(hB[i], sWih2[j * 10 + i], s);
            #pragma unroll
            for (int k = 0; k < 10; ++k) s = fmaf(hC[k], sWhh2[j * 10 + k], s);
            hn[j] = tanhf(s);
        }
        #pragma unroll
        for (int j = 0; j < 10; ++j) hC[j] = hn[j];
        // ReLU + FC accumulation: out[c] += relu(hC[j]) * fcw[c, t*10+j]
        #pragma unroll
        for (int j = 0; j < 10; ++j) {
            const float r = fmaxf(hC[j], 0.0f);
            #pragma unroll
            for (int cc = 0; cc < 5; ++cc)
                acc[cc] = fmaf(r, sfcw[cc * 500 + t * 10 + j], acc[cc]);
        }
    }

    float* op = out + (size_t)b * 5;
    #pragma unroll
    for (int cc = 0; cc < 5; ++cc) op[cc] = acc[cc];
}

extern "C" void kernel_launch(void* const* d_in, const int* in_sizes, int n_in,
                              void* d_out, int out_size, void* d_ws, size_t ws_size,
                              hipStream_t stream) {
    const float* x    = (const float*)d_in[0];   // [8192,50,50]
    const float* h0   = (const float*)d_in[1];   // [3,8192,10]
    const float* Wih0 = (const float*)d_in[2];
    const float* Whh0 = (const float*)d_in[3];
    const float* bih0 = (const float*)d_in[4];
    const float* bhh0 = (const float*)d_in[5];
    const float* Wih1 = (const float*)d_in[6];
    const float* Whh1 = (const float*)d_in[7];
    const float* bih1 = (const float*)d_in[8];
    const float* bhh1 = (const float*)d_in[9];
    const float* Wih2 = (const float*)d_in[10];
    const float* Whh2 = (const float*)d_in[11];
    const float* bih2 = (const float*)d_in[12];
    const float* bhh2 = (const float*)d_in[13];
    const float* fcw  = (const float*)d_in[14];
    const float* fcb  = (const float*)d_in[15];

    float* xw0 = (float*)d_ws;   // needs 409600*10*4 = 16.384 MB

    // 409600 rows / 16 per tile = 25600 tiles; 8 waves (tiles) per 256-thread block
    rnn_l0_proj_wmma<<<3200, 256, 0, stream>>>(x, Wih0, bih0, bhh0, xw0);
    // 8192 batch rows, 1 thread each
    rnn_recurrent_fused<<<32, 256, 0, stream>>>(xw0, h0, Whh0,
                                                Wih1, Whh1, bih1, bhh1,
                                                Wih2, Whh2, bih2, bhh2,
                                                fcw, fcb, (float*)d_out);
}